// TransformerConvBlock_5781025980850
// MI455X (gfx1250) — compile-verified
//
#include <hip/hip_runtime.h>
#include <hip/hip_bf16.h>

#define BB 4
#define NN 1024
#define EE 2048
#define DD 256
#define HH 8
#define DH 32

typedef __attribute__((ext_vector_type(2))) float v2f;
typedef __attribute__((ext_vector_type(8))) float v8f;

// ---------------------------------------------------------------------------
// fp32 WMMA: D(16x16) = A(16x4) * B(4x16) + C   (exact fp32 accumulate)
// A frag per lane: lanes 0-15 hold row M=lane, K={0,1}; lanes 16-31 K={2,3}.
// B frag per lane: lanes 0-15 hold col N=lane, K={0,1}; lanes 16-31 K={2,3}.
// C/D: VGPR r -> row r (lanes 0-15) / row r+8 (lanes 16-31), col = lane&15.
// ---------------------------------------------------------------------------
__device__ __forceinline__ v8f wmma4(v2f a, v2f b, v8f c) {
  return __builtin_amdgcn_wmma_f32_16x16x4_f32(
      /*neg_a=*/false, a, /*neg_b=*/false, b,
      /*c_mod=*/(short)0, c, /*reuse_a=*/false, /*reuse_b=*/false);
}

// reductions across a 16-lane half of the wave (rows of a C-layout tile)
__device__ __forceinline__ float rmax16(float v) {
  v = fmaxf(v, __shfl_xor(v, 1, 32));
  v = fmaxf(v, __shfl_xor(v, 2, 32));
  v = fmaxf(v, __shfl_xor(v, 4, 32));
  v = fmaxf(v, __shfl_xor(v, 8, 32));
  return v;
}
__device__ __forceinline__ float rsum16(float v) {
  v += __shfl_xor(v, 1, 32);
  v += __shfl_xor(v, 2, 32);
  v += __shfl_xor(v, 4, 32);
  v += __shfl_xor(v, 8, 32);
  return v;
}
__device__ __forceinline__ float rsum32(float v) {
  v += __shfl_xor(v, 1, 32);
  v += __shfl_xor(v, 2, 32);
  v += __shfl_xor(v, 4, 32);
  v += __shfl_xor(v, 8, 32);
  v += __shfl_xor(v, 16, 32);
  return v;
}

// ---------------------------------------------------------------------------
// LayerNorm over 256 features: one wave per row, 8 elems/lane.
// Optional second input (X2) added before normalizing (for LN3 on v + attn).
// ---------------------------------------------------------------------------
__global__ void ln_kernel(const float* __restrict__ X, int ldx, int offx,
                          const float* __restrict__ X2, int ldx2, int offx2,
                          const float* __restrict__ g, const float* __restrict__ bta,
                          float* __restrict__ Y, int rows) {
  int wave = (blockIdx.x * blockDim.x + threadIdx.x) >> 5;
  int lane = threadIdx.x & 31;
  if (wave >= rows) return;
  const float* xr = X + (size_t)wave * ldx + offx;
  float v[8];
#pragma unroll
  for (int i = 0; i < 8; ++i) v[i] = xr[lane + 32 * i];
  if (X2) {
    const float* x2r = X2 + (size_t)wave * ldx2 + offx2;
#pragma unroll
    for (int i = 0; i < 8; ++i) v[i] += x2r[lane + 32 * i];
  }
  float s = 0.f;
#pragma unroll
  for (int i = 0; i < 8; ++i) s += v[i];
  s = rsum32(s);
  float mean = s * (1.0f / 256.0f);
  float q = 0.f;
#pragma unroll
  for (int i = 0; i < 8; ++i) { float d = v[i] - mean; q += d * d; }
  q = rsum32(q);
  float inv = rsqrtf(q * (1.0f / 256.0f) + 1e-5f);
  float* yr = Y + (size_t)wave * DD;
#pragma unroll
  for (int i = 0; i < 8; ++i) {
    int c = lane + 32 * i;
    yr[c] = (v[i] - mean) * inv * g[c] + bta[c];
  }
}

// ---------------------------------------------------------------------------
// Build gather row maps:  srow[b*E+e] = b*N + edge_index[b,0,e], drow = [b,1,e]
// ---------------------------------------------------------------------------
__global__ void rows_kernel(const int* __restrict__ ei, int* __restrict__ srow,
                            int* __restrict__ drow) {
  int i = blockIdx.x * blockDim.x + threadIdx.x;
  if (i >= BB * EE) return;
  int b = i / EE, e = i % EE;
  srow[i] = b * NN + ei[(size_t)b * 2 * EE + e];
  drow[i] = b * NN + ei[(size_t)b * 2 * EE + EE + e];
}

// ---------------------------------------------------------------------------
// Generic fp32 WMMA GEMM:  C[m, n] = A[rows(m), :] @ W + bias (+ A2 @ W2 + bias2)
// One wave computes a 16x64 output tile (4 accumulators sharing the A frag).
// M = 8192 rows fixed by launch; N, K passed in. a_rows = optional gather.
// ---------------------------------------------------------------------------
__global__ void wmma_gemm(const float* __restrict__ A, const int* __restrict__ a_rows,
                          int lda, const float* __restrict__ W,
                          const float* __restrict__ bias,
                          const float* __restrict__ A2, const float* __restrict__ W2,
                          const float* __restrict__ bias2,
                          float* __restrict__ C, int ldc, int N, int K) {
  int wid  = (blockIdx.x * blockDim.x + threadIdx.x) >> 5;
  int lane = threadIdx.x & 31;
  int half = lane >> 4, lm = lane & 15, kh = half * 2;
  int nt64 = N >> 6;
  int mt = wid / nt64, nt = wid % nt64;
  int m0 = mt * 16, n0b = nt * 64;

  v8f acc[4];
#pragma unroll
  for (int t = 0; t < 4; ++t) acc[t] = (v8f){};

  int r = m0 + lm;
  int arow = a_rows ? a_rows[r] : r;
  const float* ap_base = A + (size_t)arow * lda + kh;

#pragma unroll 4
  for (int k0 = 0; k0 < K; k0 += 4) {
    v2f a;
    a.x = ap_base[k0];
    a.y = ap_base[k0 + 1];
    const float* wp = W + (size_t)(k0 + kh) * N + n0b + lm;
#pragma unroll
    for (int t = 0; t < 4; ++t) {
      v2f b;
      b.x = wp[t * 16];
      b.y = wp[t * 16 + N];
      acc[t] = wmma4(a, b, acc[t]);
    }
  }

  if (A2) {  // fused second accumulation (e_ln @ we), identity row map
    const float* ap2 = A2 + (size_t)r * lda + kh;
#pragma unroll 4
    for (int k0 = 0; k0 < K; k0 += 4) {
      v2f a;
      a.x = ap2[k0];
      a.y = ap2[k0 + 1];
      const float* wp = W2 + (size_t)(k0 + kh) * N + n0b + lm;
#pragma unroll
      for (int t = 0; t < 4; ++t) {
        v2f b;
        b.x = wp[t * 16];
        b.y = wp[t * 16 + N];
        acc[t] = wmma4(a, b, acc[t]);
      }
    }
  }

#pragma unroll
  for (int t = 0; t < 4; ++t) {
    int col = n0b + t * 16 + lm;
    float bv = bias ? bias[col] : 0.f;
    if (bias2) bv += bias2[col];
#pragma unroll
    for (int rr = 0; rr < 8; ++rr) {
      int orow = m0 + rr + half * 8;
      C[(size_t)orow * ldc + col] = acc[t][rr] + bv;
    }
  }
}

// ---------------------------------------------------------------------------
// Flash attention over E=2048 keys per (b, h). One wave owns a 16-query tile.
// Online softmax; P transposed C->A layout through LDS; all MACs via WMMA.
// Q: (B*E, 256). KV: (B*E, 512), k = cols [0,256), v = cols [256,512).
// ---------------------------------------------------------------------------
__global__ void attn_kernel(const float* __restrict__ Q, const float* __restrict__ KV,
                            float* __restrict__ Oa) {
  __shared__ float plds[4][16][18];
  int b = blockIdx.z, h = blockIdx.y;
  int wv = threadIdx.x >> 5, lane = threadIdx.x & 31;
  int half = lane >> 4, lm = lane & 15, kh = half * 2;
  int e0 = (blockIdx.x * 4 + wv) * 16;
  int hoff = h * DH;
  const float scale = 0.17677669529663687f;  // 1/sqrt(32)

  // load Q A-fragments (16 rows x dh=32 -> 8 K-chunks)
  v2f qa[8];
  const float* qrow = Q + ((size_t)b * EE + e0 + lm) * DD + hoff + kh;
#pragma unroll
  for (int kk = 0; kk < 8; ++kk) { qa[kk].x = qrow[4 * kk]; qa[kk].y = qrow[4 * kk + 1]; }

  float mrow[8], lrow[8];
  v8f acc0 = (v8f){}, acc1 = (v8f){};
#pragma unroll
  for (int rr = 0; rr < 8; ++rr) { mrow[rr] = -3.0e38f; lrow[rr] = 0.f; }

  for (int j = 0; j < EE / 16; ++j) {
    size_t kb = (size_t)b * EE + j * 16;
    // S = Q @ K^T  (B frag: col n = key lane, contiguous feature pair)
    v8f s = (v8f){};
    const float* kp = KV + (kb + lm) * 512 + hoff + kh;
#pragma unroll
    for (int kk = 0; kk < 8; ++kk) {
      v2f bf;
      bf.x = kp[4 * kk];
      bf.y = kp[4 * kk + 1];
      s = wmma4(qa[kk], bf, s);
    }
    // online softmax on the 16x16 score tile
#pragma unroll
    for (int rr = 0; rr < 8; ++rr) {
      float sv = s[rr] * scale;
      float mnew = fmaxf(mrow[rr], rmax16(sv));
      float alpha = __expf(mrow[rr] - mnew);
      float p = __expf(sv - mnew);
      lrow[rr] = lrow[rr] * alpha + rsum16(p);
      mrow[rr] = mnew;
      acc0[rr] *= alpha;
      acc1[rr] *= alpha;
      plds[wv][rr + half * 8][lm] = p;  // C-layout -> row-major stash
    }
    asm volatile("s_wait_dscnt 0" ::: "memory");  // wave-local LDS RAW
    // O += P @ V   (A frag of P from LDS; B frag = V rows, two 16-col tiles)
#pragma unroll
    for (int kk = 0; kk < 4; ++kk) {
      v2f pa;
      pa.x = plds[wv][lm][4 * kk + kh];
      pa.y = plds[wv][lm][4 * kk + kh + 1];
      const float* vp = KV + (kb + 4 * kk + kh) * 512 + 256 + hoff + lm;
      v2f b0, b1;
      b0.x = vp[0];    b0.y = vp[512];
      b1.x = vp[16];   b1.y = vp[512 + 16];
      acc0 = wmma4(pa, b0, acc0);
      acc1 = wmma4(pa, b1, acc1);
    }
  }
#pragma unroll
  for (int rr = 0; rr < 8; ++rr) {
    float invl = 1.0f / lrow[rr];
    float* op = Oa + ((size_t)b * EE + e0 + rr + half * 8) * DD + hoff;
    op[lm]      = acc0[rr] * invl;
    op[16 + lm] = acc1[rr] * invl;
  }
}

// ---------------------------------------------------------------------------
// Fused FFN: 16-row strip per workgroup; 16x1024 hidden kept entirely in LDS
// (64 KB of the WGP's 320 KB). Out = relu(H@w1+b1)@w2 + b2 + H.
// ---------------------------------------------------------------------------
__global__ void ffn_kernel(const float* __restrict__ H, const float* __restrict__ w1,
                           const float* __restrict__ b1, const float* __restrict__ w2,
                           const float* __restrict__ b2, float* __restrict__ Out) {
  __shared__ float h1s[16 * 1024];
  int m0 = blockIdx.x * 16;
  int wv = threadIdx.x >> 5, lane = threadIdx.x & 31;
  int half = lane >> 4, lm = lane & 15, kh = half * 2;

  // phase 1: hidden strip = relu(H @ w1 + b1), wave wv owns cols [wv*128, +128)
  const float* ap = H + (size_t)(m0 + lm) * DD + kh;
#pragma unroll
  for (int t = 0; t < 8; ++t) {
    int n0 = wv * 128 + t * 16;
    v8f acc = (v8f){};
#pragma unroll 4
    for (int k0 = 0; k0 < 256; k0 += 4) {
      v2f a; a.x = ap[k0]; a.y = ap[k0 + 1];
      const float* wp = w1 + (size_t)(k0 + kh) * 1024 + n0 + lm;
      v2f b; b.x = wp[0]; b.y = wp[1024];
      acc = wmma4(a, b, acc);
    }
    float bv = b1[n0 + lm];
#pragma unroll
    for (int rr = 0; rr < 8; ++rr)
      h1s[(rr + half * 8) * 1024 + n0 + lm] = fmaxf(acc[rr] + bv, 0.f);
  }
  __syncthreads();

  // phase 2: Out strip, wave wv owns cols [wv*32, +32), K=1024 from LDS
#pragma unroll
  for (int t = 0; t < 2; ++t) {
    int n0 = (wv * 2 + t) * 16;
    v8f acc = (v8f){};
#pragma unroll 4
    for (int k0 = 0; k0 < 1024; k0 += 4) {
      v2f a;
      a.x = h1s[lm * 1024 + k0 + kh];
      a.y = h1s[lm * 1024 + k0 + kh + 1];
      const float* wp = w2 + (size_t)(k0 + kh) * DD + n0 + lm;
      v2f b; b.x = wp[0]; b.y = wp[DD];
      acc = wmma4(a, b, acc);
    }
    float bv = b2[n0 + lm];
#pragma unroll
    for (int rr = 0; rr < 8; ++rr) {
      int orow = m0 + rr + half * 8;
      Out[(size_t)orow * DD + n0 + lm] =
          acc[rr] + bv + H[(size_t)orow * DD + n0 + lm];
    }
  }
}

// ---------------------------------------------------------------------------
// Segment sum + count over dst (fp32 atomics into L2-resident buffers)
// ---------------------------------------------------------------------------
__global__ void seg_kernel(const float* __restrict__ hid, const int* __restrict__ ei,
                           float* __restrict__ seg, float* __restrict__ cnt) {
  int be = blockIdx.x;
  int b = be / EE, e = be % EE;
  int d = threadIdx.x;
  int dst = ei[(size_t)b * 2 * EE + EE + e];
  atomicAdd(&seg[((size_t)b * NN + dst) * DD + d], hid[(size_t)be * DD + d]);
  if (d == 0) atomicAdd(&cnt[b * NN + dst], 1.0f);
}

__global__ void final_kernel(const float* __restrict__ seg, const float* __restrict__ cnt,
                             const float* __restrict__ xln, float* __restrict__ out) {
  int i = blockIdx.x * blockDim.x + threadIdx.x;  // B*N*D
  int bn = i >> 8;
  float mean = seg[i] / fmaxf(cnt[bn], 1.0f);
  out[i] = (mean != 0.0f) ? mean : xln[i];
}

// ---------------------------------------------------------------------------
extern "C" void kernel_launch(void* const* d_in, const int* in_sizes, int n_in,
                              void* d_out, int out_size, void* d_ws, size_t ws_size,
                              hipStream_t stream) {
  (void)in_sizes; (void)n_in; (void)out_size; (void)ws_size;
  const float* x     = (const float*)d_in[0];
  const float* eattr = (const float*)d_in[1];
  const float* ln1_g = (const float*)d_in[2];
  const float* ln1_b = (const float*)d_in[3];
  const float* ln2_g = (const float*)d_in[4];
  const float* ln2_b = (const float*)d_in[5];
  const float* ln3_g = (const float*)d_in[6];
  const float* ln3_b = (const float*)d_in[7];
  const float* wq    = (const float*)d_in[8];
  const float* bq    = (const float*)d_in[9];
  const float* wkv   = (const float*)d_in[10];
  const float* bkv   = (const float*)d_in[11];
  const float* we    = (const float*)d_in[12];
  const float* be    = (const float*)d_in[13];
  const float* w1    = (const float*)d_in[14];
  const float* b1    = (const float*)d_in[15];
  const float* w2    = (const float*)d_in[16];
  const float* b2    = (const float*)d_in[17];
  const int*   ei    = (const int*)d_in[18];

  float* ws   = (float*)d_ws;
  float* x_ln = ws;                  // B*N*D   = 1048576
  float* e_ln = x_ln + 1048576;      // B*E*D   = 2097152
  float* qb   = e_ln + 2097152;      // B*E*D   = 2097152
  float* kvb  = qb   + 2097152;      // B*E*2D  = 4194304
  float* attn = kvb  + 4194304;      // B*E*D   = 2097152
  float* hln  = attn + 2097152;      // B*E*D   = 2097152
  float* ffo  = hln  + 2097152;      // B*E*D   = 2097152
  float* seg  = ffo  + 2097152;      // B*N*D   = 1048576
  float* cnt  = seg  + 1048576;      // B*N     = 4096
  int*   srow = (int*)(cnt + 4096);  // B*E
  int*   drow = srow + BB * EE;      // B*E

  // 1) LayerNorms
  ln_kernel<<<512, 256, 0, stream>>>(x, DD, 0, nullptr, 0, 0, ln1_g, ln1_b, x_ln, BB * NN);
  ln_kernel<<<1024, 256, 0, stream>>>(eattr, DD, 0, nullptr, 0, 0, ln2_g, ln2_b, e_ln, BB * EE);

  // 2) gather row maps
  rows_kernel<<<(BB * EE + 255) / 256, 256, 0, stream>>>(ei, srow, drow);

  // 3) q = x_ln[dst] @ wq + bq + e_ln @ we + be    (2048 wave-tiles)
  wmma_gemm<<<256, 256, 0, stream>>>(x_ln, drow, DD, wq, bq, e_ln, we, be, qb, DD, DD, DD);

  // 4) kv = x_ln[src] @ wkv + bkv                  (4096 wave-tiles)
  wmma_gemm<<<512, 256, 0, stream>>>(x_ln, srow, DD, wkv, bkv, nullptr, nullptr, nullptr,
                                     kvb, 2 * DD, 2 * DD, DD);

  // 5) flash attention per (b, h)
  attn_kernel<<<dim3(EE / 64, HH, BB), 128, 0, stream>>>(qb, kvb, attn);

  // 6) hidden = LN(v + a)
  ln_kernel<<<1024, 256, 0, stream>>>(attn, DD, 0, kvb, 2 * DD, DD, ln3_g, ln3_b, hln, BB * EE);

  // 7) fused FFN with hidden strip in LDS
  ffn_kernel<<<BB * EE / 16, 256, 0, stream>>>(hln, w1, b1, w2, b2, ffo);

  // 8) segment mean over dst, then select
  hipMemsetAsync(seg, 0, (size_t)(1048576 + 4096) * sizeof(float), stream);
  seg_kernel<<<BB * EE, 256, 0, stream>>>(ffo, ei, seg, cnt);
  final_kernel<<<BB * NN * DD / 256, 256, 0, stream>>>(seg, cnt, x_ln, (float*)d_out);
}